// AttentionMechanism_30537217475345
// MI455X (gfx1250) — compile-verified
//
#include <hip/hip_runtime.h>
#include <hip/hip_bf16.h>

// Sizes from the reference
#define BATCH 32
#define LSEQ  2048
#define KDIM  512
#define ADIM  512
#define COUT  10
#define KHALF 100
#define KTAPS 201
#define NEGV  (-1024.0f)

typedef __attribute__((ext_vector_type(2))) float v2f;
typedef __attribute__((ext_vector_type(4))) float v4f;
typedef __attribute__((ext_vector_type(8))) float v8f;

// ---------------------------------------------------------------------------
// q_proj[b][a] = sum_k query[b][k] * w_query[k][a]
// ---------------------------------------------------------------------------
__global__ __launch_bounds__(256) void qproj_kernel(const float* __restrict__ query,
                                                    const float* __restrict__ w_query,
                                                    float* __restrict__ qproj) {
    int idx = blockIdx.x * 256 + threadIdx.x;        // BATCH*ADIM = 16384
    int b = idx >> 9;
    int a = idx & (ADIM - 1);
    const float* q = query + b * KDIM;
    float s = 0.f;
    #pragma unroll 8
    for (int k = 0; k < KDIM; ++k) s = fmaf(q[k], w_query[k * ADIM + a], s);
    qproj[idx] = s;
}

// ---------------------------------------------------------------------------
// conv_feat[b][c][l] = sum_t aw[b][l-100+t] * conv_w[c][t]  (same padding)
// ---------------------------------------------------------------------------
__global__ __launch_bounds__(256) void conv_kernel(const float* __restrict__ aw,
                                                   const float* __restrict__ conv_w,
                                                   float* __restrict__ cf) {
    int idx = blockIdx.x * 256 + threadIdx.x;        // BATCH*COUT*LSEQ = 655360
    int l = idx & (LSEQ - 1);
    int c = (idx >> 11) % COUT;
    int b = idx / (LSEQ * COUT);
    const float* awb = aw + (size_t)b * LSEQ;
    const float* w   = conv_w + c * KTAPS;
    float s = 0.f;
    for (int t = 0; t < KTAPS; ++t) {
        int p = l - KHALF + t;
        float av = (p >= 0 && p < LSEQ) ? awb[p] : 0.f;
        s = fmaf(av, w[t], s);
    }
    cf[idx] = s;                                     // layout [b][c][l]
}

// ---------------------------------------------------------------------------
// Energy kernel (WMMA): e[b][l] = v . tanh(key@w_key + b_key + qproj + cproj)
// One block = 16 rows of one batch. 8 waves; wave w owns N-tiles {w, w+8, w+16, w+24}.
// ---------------------------------------------------------------------------
#define LDS_STRIDE 516   // 512 + 4 pad floats -> bank rotation, keeps 16B alignment

__global__ __launch_bounds__(256) void energy_kernel(const float* __restrict__ key,
                                                     const float* __restrict__ w_key,
                                                     const float* __restrict__ b_key,
                                                     const float* __restrict__ vvec,
                                                     const float* __restrict__ w_conv,
                                                     const float* __restrict__ qproj,
                                                     const float* __restrict__ cf,
                                                     float* __restrict__ e_out) {
    __shared__ float sKey[16 * LDS_STRIDE];
    __shared__ float sCF[16][COUT];
    __shared__ float sE[16];

    const int b     = blockIdx.x >> 7;               // / (LSEQ/16)
    const int lbase = (blockIdx.x & 127) << 4;
    const int tid   = threadIdx.x;

    // Stage 16x512 key tile (streamed once from HBM -> nontemporal)
    const v4f* gk = (const v4f*)(key + ((size_t)b * LSEQ + lbase) * KDIM);
    for (int i = tid; i < 16 * 128; i += 256) {      // 2048 float4
        int r = i >> 7, c4 = i & 127;
        v4f val = __builtin_nontemporal_load(gk + r * 128 + c4);
        float* dst = &sKey[r * LDS_STRIDE + c4 * 4];
        dst[0] = val.x; dst[1] = val.y; dst[2] = val.z; dst[3] = val.w;
    }
    if (tid < 16 * COUT) {
        int r = tid / COUT, c = tid % COUT;
        sCF[r][c] = cf[((size_t)b * COUT + c) * LSEQ + lbase + r];
    }
    if (tid < 16) sE[tid] = 0.f;
    __syncthreads();

    const int wave = tid >> 5;
    const int lane = tid & 31;
    const int half = lane >> 4;                      // K sub-pair select (A), row+8 (C/D)
    const int ln   = lane & 15;                      // row (A) / col (B,C,D)

    v8f acc0 = {}, acc1 = {}, acc2 = {}, acc3 = {};
    const int nb0 = (wave +  0) << 4;
    const int nb1 = (wave +  8) << 4;
    const int nb2 = (wave + 16) << 4;
    const int nb3 = (wave + 24) << 4;

    // A-matrix 16x4 f32 layout: vgpr0 = A[ln][k+2*half], vgpr1 = A[ln][k+2*half+1]
    const float* sA = &sKey[ln * LDS_STRIDE + 2 * half];

    #pragma unroll 2
    for (int k = 0; k < KDIM; k += 4) {
        v2f a;
        a.x = sA[k];
        a.y = sA[k + 1];
        const float* wrow = w_key + (size_t)(k + 2 * half) * ADIM + ln;
        v2f bf0, bf1, bf2, bf3;
        bf0.x = wrow[nb0]; bf0.y = wrow[ADIM + nb0];
        bf1.x = wrow[nb1]; bf1.y = wrow[ADIM + nb1];
        bf2.x = wrow[nb2]; bf2.y = wrow[ADIM + nb2];
        bf3.x = wrow[nb3]; bf3.y = wrow[ADIM + nb3];
        acc0 = __builtin_amdgcn_wmma_f32_16x16x4_f32(false, a, false, bf0, (short)0, acc0, false, false);
        acc1 = __builtin_amdgcn_wmma_f32_16x16x4_f32(false, a, false, bf1, (short)0, acc1, false, false);
        acc2 = __builtin_amdgcn_wmma_f32_16x16x4_f32(false, a, false, bf2, (short)0, acc2, false, false);
        acc3 = __builtin_amdgcn_wmma_f32_16x16x4_f32(false, a, false, bf3, (short)0, acc3, false, false);
    }

    // Epilogue: + b_key + qproj + conv-proj, tanh, * v, reduce over columns
    const float* qpb = qproj + b * ADIM;
    v8f accs[4] = {acc0, acc1, acc2, acc3};
    int  nbs[4] = {nb0, nb1, nb2, nb3};
    #pragma unroll
    for (int j = 0; j < 4; ++j) {
        int col = nbs[j] + ln;
        float qv = qpb[col] + b_key[col];
        float vv = vvec[col];
        float wc[COUT];
        #pragma unroll
        for (int c = 0; c < COUT; ++c) wc[c] = w_conv[c * ADIM + col];
        #pragma unroll
        for (int i = 0; i < 8; ++i) {
            int row = i + 8 * half;                  // C/D layout: M = i + 8*(lane>=16)
            float s = accs[j][i] + qv;
            #pragma unroll
            for (int c = 0; c < COUT; ++c) s = fmaf(sCF[row][c], wc[c], s);
            float t = tanhf(s) * vv;
            // sum across the 16 columns held by this lane-half
            t += __shfl_xor(t, 1);
            t += __shfl_xor(t, 2);
            t += __shfl_xor(t, 4);
            t += __shfl_xor(t, 8);
            if (ln == 0) atomicAdd(&sE[row], t);     // ds_add_f32
        }
    }
    __syncthreads();
    if (tid < 16) e_out[(size_t)b * LSEQ + lbase + tid] = sE[tid];
}

// ---------------------------------------------------------------------------
// Masked softmax over L per batch; writes aw_new to d_out and zeroes cv region
// ---------------------------------------------------------------------------
__global__ __launch_bounds__(256) void softmax_kernel(const float* __restrict__ e,
                                                      const int* __restrict__ klens,
                                                      float* __restrict__ out) {
    __shared__ float red[256];
    const int b = blockIdx.x, tid = threadIdx.x;
    const int klen = klens[b];
    const float* eb = e + (size_t)b * LSEQ;

    float ev[8];
    float mx = -3.4e38f;
    #pragma unroll
    for (int j = 0; j < 8; ++j) {
        int l = tid + j * 256;
        float x = (l < klen) ? eb[l] : NEGV;
        ev[j] = x;
        mx = fmaxf(mx, x);
    }
    red[tid] = mx; __syncthreads();
    for (int s = 128; s > 0; s >>= 1) { if (tid < s) red[tid] = fmaxf(red[tid], red[tid + s]); __syncthreads(); }
    mx = red[0]; __syncthreads();

    float pv[8], sum = 0.f;
    #pragma unroll
    for (int j = 0; j < 8; ++j) { pv[j] = __expf(ev[j] - mx); sum += pv[j]; }
    red[tid] = sum; __syncthreads();
    for (int s = 128; s > 0; s >>= 1) { if (tid < s) red[tid] += red[tid + s]; __syncthreads(); }
    float inv = 1.f / red[0];

    float* awn = out + BATCH * ADIM;                 // aw_new region after cv
    #pragma unroll
    for (int j = 0; j < 8; ++j) awn[(size_t)b * LSEQ + tid + j * 256] = pv[j] * inv;

    // zero cv region (d_out is poisoned; context kernel atomically accumulates)
    out[b * ADIM + tid]       = 0.f;
    out[b * ADIM + 256 + tid] = 0.f;
}

// ---------------------------------------------------------------------------
// cv[b][d] = sum_l aw_new[b][l] * value[b][l][d]   (grid (B, 8), 512 threads)
// ---------------------------------------------------------------------------
__global__ __launch_bounds__(512) void context_kernel(const float* __restrict__ value,
                                                      const float* __restrict__ aw_new,
                                                      float* __restrict__ cv) {
    __shared__ float sP[256];
    const int b = blockIdx.x, seg = blockIdx.y, tid = threadIdx.x;
    const int l0 = seg * 256;
    if (tid < 256) sP[tid] = aw_new[(size_t)b * LSEQ + l0 + tid];
    __syncthreads();
    const float* vb = value + ((size_t)b * LSEQ + l0) * KDIM + tid;
    float acc = 0.f;
    #pragma unroll 4
    for (int i = 0; i < 256; ++i)
        acc = fmaf(sP[i], __builtin_nontemporal_load(vb + (size_t)i * KDIM), acc);
    atomicAdd(&cv[b * ADIM + tid], acc);
}

// ---------------------------------------------------------------------------
extern "C" void kernel_launch(void* const* d_in, const int* in_sizes, int n_in,
                              void* d_out, int out_size, void* d_ws, size_t ws_size,
                              hipStream_t stream) {
    const float* key     = (const float*)d_in[0];
    const int*   klens   = (const int*)  d_in[1];
    const float* value   = (const float*)d_in[2];
    const float* query   = (const float*)d_in[3];
    const float* aw      = (const float*)d_in[4];
    const float* w_key   = (const float*)d_in[5];
    const float* b_key   = (const float*)d_in[6];
    const float* w_query = (const float*)d_in[7];
    const float* w_conv  = (const float*)d_in[8];
    const float* conv_w  = (const float*)d_in[9];
    const float* v       = (const float*)d_in[10];

    float* out = (float*)d_out;
    float* ws  = (float*)d_ws;
    float* qproj = ws;                                   // 16384 floats
    float* cf    = ws + BATCH * ADIM;                    // 655360 floats
    float* e     = ws + BATCH * ADIM + BATCH * COUT * LSEQ; // 65536 floats

    qproj_kernel <<<(BATCH * ADIM) / 256, 256, 0, stream>>>(query, w_query, qproj);
    conv_kernel  <<<(BATCH * COUT * LSEQ) / 256, 256, 0, stream>>>(aw, conv_w, cf);
    energy_kernel<<<BATCH * (LSEQ / 16), 256, 0, stream>>>(key, w_key, b_key, v,
                                                           w_conv, qproj, cf, e);
    softmax_kernel<<<BATCH, 256, 0, stream>>>(e, klens, out);
    context_kernel<<<dim3(BATCH, 8), 512, 0, stream>>>(value, out + BATCH * ADIM, out);
}